// GNN_7456063225891
// MI455X (gfx1250) — compile-verified
//
#include <hip/hip_runtime.h>
#include <hip/hip_bf16.h>

#define N_NODES 100000
#define N_EDGES 1600000
#define N_GRAPHS 1024
#define NODE_F 100
#define HID 64
#define GLOB_F 3

typedef float v2f __attribute__((ext_vector_type(2)));
typedef float v8f __attribute__((ext_vector_type(8)));

__device__ __forceinline__ void atomic_add_f32(float* p, float v) {
  unsafeAtomicAdd(p, v);  // lowers to global_atomic_add_f32 (no CAS loop)
}

// ---------------- setup: deg=1 (self-loop), pool sums/counts = 0 ----------
__global__ void k_setup(float* deg, float* psum, float* pcnt) {
  int i = blockIdx.x * blockDim.x + threadIdx.x;
  if (i < N_NODES) deg[i] = 1.0f;
  if (i < N_GRAPHS * HID) psum[i] = 0.0f;
  if (i < N_GRAPHS) pcnt[i] = 0.0f;
}

__global__ void k_deg(const int* __restrict__ dst, float* deg) {
  int e = blockIdx.x * blockDim.x + threadIdx.x;
  if (e < N_EDGES) atomic_add_f32(&deg[dst[e]], 1.0f);
}

__global__ void k_dinv(float* deg) {
  int i = blockIdx.x * blockDim.x + threadIdx.x;
  if (i < N_NODES) deg[i] = rsqrtf(deg[i]);   // in place: deg -> d^-1/2
}

// ---------------- WMMA fp32 GEMM: out[M x 64] = A[M x K] @ W[K x 64] ------
// One wave computes a 16x64 tile: 4 accumulators, K/4 wmma steps each.
// A frag (16x4 f32): lane r=lane&15 row, koff=2*(lane>>4): {A[r][k+koff], A[r][k+koff+1]}
// B frag (4x16 f32): {W[kk][n], W[kk+1][n]}, kk = k+koff, n = tile*16 + r
//   -> W staged in LDS k-pair-interleaved: sW2[(kk/2)*PADW + n] = {W[kk][n], W[kk+1][n]}
//      single ds_load_b64 per fragment, PADW=80 keeps half-wave lane groups on
//      disjoint 32-bank halves (2*80 mod 64 == 32).
// D (16x16 f32): VGPR j -> row j + 8*(lane>>4), col r
#define PADW 80

template <int K, bool RELU_A>
__global__ void k_gemm_wmma(const float* __restrict__ A,
                            const float* __restrict__ W,
                            float* __restrict__ out, int M) {
  __shared__ v2f sW2[(K / 2) * PADW];
  int tid = threadIdx.x;
  for (int i = tid; i < (K / 2) * HID; i += blockDim.x) {
    int kp = i >> 6;            // k pair index
    int n  = i & 63;
    v2f w; w.x = W[(2 * kp) * HID + n]; w.y = W[(2 * kp + 1) * HID + n];
    sW2[kp * PADW + n] = w;
  }
  __syncthreads();

  int wave = tid >> 5;
  int lane = tid & 31;
  int row16 = (blockIdx.x * 8 + wave) * 16;
  if (row16 >= M) return;                 // M % 16 == 0, tiles never split

  int r = lane & 15;
  int hi = lane >> 4;                     // 0 or 1
  int koff = hi * 2;
  const float* arow = A + (size_t)(row16 + r) * K;

  v8f acc0 = {}, acc1 = {}, acc2 = {}, acc3 = {};
#pragma unroll 4
  for (int k = 0; k < K; k += 4) {
    float a0 = arow[k + koff];
    float a1 = arow[k + koff + 1];
    if (RELU_A) { a0 = fmaxf(a0, 0.0f); a1 = fmaxf(a1, 0.0f); }
    v2f a; a.x = a0; a.y = a1;
    const v2f* wrow = &sW2[(k / 2 + hi) * PADW + r];
    v2f b0 = wrow[0];
    v2f b1 = wrow[16];
    v2f b2 = wrow[32];
    v2f b3 = wrow[48];
    acc0 = __builtin_amdgcn_wmma_f32_16x16x4_f32(false, a, false, b0, (short)0, acc0, false, false);
    acc1 = __builtin_amdgcn_wmma_f32_16x16x4_f32(false, a, false, b1, (short)0, acc1, false, false);
    acc2 = __builtin_amdgcn_wmma_f32_16x16x4_f32(false, a, false, b2, (short)0, acc2, false, false);
    acc3 = __builtin_amdgcn_wmma_f32_16x16x4_f32(false, a, false, b3, (short)0, acc3, false, false);
  }

  int rowhi = hi * 8;
#pragma unroll
  for (int j = 0; j < 8; j++) {
    size_t o = (size_t)(row16 + j + rowhi) * HID + r;
    out[o +  0] = acc0[j];
    out[o + 16] = acc1[j];
    out[o + 32] = acc2[j];
    out[o + 48] = acc3[j];
  }
}

// ---------------- aggregation: out = scatter(h[src]*norm) + selfloop + b ---
__global__ void k_agg_init(const float* __restrict__ t, const float* __restrict__ dinv,
                           const float* __restrict__ b, float* __restrict__ out) {
  int idx = blockIdx.x * blockDim.x + threadIdx.x;
  if (idx >= N_NODES * HID) return;
  int n = idx >> 6, c = idx & 63;
  float di = dinv[n];
  out[idx] = t[idx] * di * di + b[c];     // self-loop term + bias
}

// wave per edge: 32 lanes x float2 columns, coalesced L2 gather + atomics
__global__ void k_agg_edges(const float* __restrict__ t, const int* __restrict__ src,
                            const int* __restrict__ dst, const float* __restrict__ dinv,
                            float* __restrict__ out) {
  int gid = blockIdx.x * blockDim.x + threadIdx.x;
  int e = gid >> 5;
  int lane = gid & 31;
  if (e >= N_EDGES) return;
  int s = src[e], d = dst[e];
  float norm = dinv[s] * dinv[d];
  float2 v = ((const float2*)(t + (size_t)s * HID))[lane];
  float* orow = out + (size_t)d * HID + 2 * lane;
  atomic_add_f32(orow + 0, v.x * norm);
  atomic_add_f32(orow + 1, v.y * norm);
}

// ---------------- global mean pool accumulation ---------------------------
__global__ void k_pool(const float* __restrict__ h, const int* __restrict__ batch,
                       float* psum, float* pcnt) {
  int gid = blockIdx.x * blockDim.x + threadIdx.x;
  int n = gid >> 5;
  int lane = gid & 31;
  if (n >= N_NODES) return;
  int g = batch[n];
  float2 v = ((const float2*)(h + (size_t)n * HID))[lane];
  float* prow = psum + (size_t)g * HID + 2 * lane;
  atomic_add_f32(prow + 0, v.x);
  atomic_add_f32(prow + 1, v.y);
  if (lane == 0) atomic_add_f32(&pcnt[g], 1.0f);
}

// ---------------- per-graph head MLPs (tiny: 1024 graphs) -----------------
__global__ void k_head(const float* __restrict__ psum, const float* __restrict__ pcnt,
                       const float* __restrict__ gf,
                       const float* __restrict__ gW1, const float* __restrict__ gb1,
                       const float* __restrict__ gW2, const float* __restrict__ gb2,
                       const float* __restrict__ cW1, const float* __restrict__ cb1,
                       const float* __restrict__ cW2, const float* __restrict__ cb2,
                       float* __restrict__ out) {
  int g = blockIdx.x * blockDim.x + threadIdx.x;
  if (g >= N_GRAPHS) return;
  float inv = 1.0f / fmaxf(pcnt[g], 1.0f);
  float gf0 = gf[g * GLOB_F + 0], gf1 = gf[g * GLOB_F + 1], gf2 = gf[g * GLOB_F + 2];
  float g1[HID];
#pragma unroll 8
  for (int c = 0; c < HID; c++)
    g1[c] = fmaxf(gf0 * gW1[c] + gf1 * gW1[HID + c] + gf2 * gW1[2 * HID + c] + gb1[c], 0.0f);
  float g2[HID];
  for (int c = 0; c < HID; c++) {
    float s = gb2[c];
    for (int k = 0; k < HID; k++) s += g1[k] * gW2[k * HID + c];
    g2[c] = s;
  }
  const float* prow = psum + (size_t)g * HID;
  float o = cb2[0];
  for (int c = 0; c < HID; c++) {
    float s = cb1[c];
    for (int k = 0; k < HID; k++) s += (prow[k] * inv) * cW1[k * HID + c];
    for (int k = 0; k < HID; k++) s += g2[k] * cW1[(HID + k) * HID + c];
    o += fmaxf(s, 0.0f) * cW2[c];
  }
  out[g] = o;
}

extern "C" void kernel_launch(void* const* d_in, const int* in_sizes, int n_in,
                              void* d_out, int out_size, void* d_ws, size_t ws_size,
                              hipStream_t stream) {
  const float* x    = (const float*)d_in[0];
  const int*   ei   = (const int*)d_in[1];
  const int*   src  = ei;
  const int*   dst  = ei + N_EDGES;
  const int*   batch = (const int*)d_in[2];
  const float* gf   = (const float*)d_in[3];
  const float* W1   = (const float*)d_in[4];
  const float* b1   = (const float*)d_in[5];
  const float* W2   = (const float*)d_in[6];
  const float* b2   = (const float*)d_in[7];
  const float* gW1  = (const float*)d_in[8];
  const float* gb1  = (const float*)d_in[9];
  const float* gW2  = (const float*)d_in[10];
  const float* gb2  = (const float*)d_in[11];
  const float* cW1  = (const float*)d_in[12];
  const float* cb1  = (const float*)d_in[13];
  const float* cW2  = (const float*)d_in[14];
  const float* cb2  = (const float*)d_in[15];
  float* out = (float*)d_out;

  // workspace layout (floats)
  float* ws   = (float*)d_ws;
  float* bufA = ws;                              // [N_NODES*HID]
  float* bufB = bufA + (size_t)N_NODES * HID;    // [N_NODES*HID]
  float* dinv = bufB + (size_t)N_NODES * HID;    // [N_NODES]
  float* psum = dinv + N_NODES;                  // [N_GRAPHS*HID]
  float* pcnt = psum + (size_t)N_GRAPHS * HID;   // [N_GRAPHS]

  const int T = 256;
  // degrees + pool init
  k_setup<<<(N_NODES + T - 1) / T, T, 0, stream>>>(dinv, psum, pcnt);
  k_deg<<<(N_EDGES + T - 1) / T, T, 0, stream>>>(dst, dinv);
  k_dinv<<<(N_NODES + T - 1) / T, T, 0, stream>>>(dinv);

  int gemm_blocks = (N_NODES / 16 + 7) / 8;      // 8 waves (16-row tiles) per block

  // conv1: t1 = x @ W1 ; a1 = aggregate(t1) + b1  (relu deferred into gemm2)
  k_gemm_wmma<NODE_F, false><<<gemm_blocks, T, 0, stream>>>(x, W1, bufA, N_NODES);
  k_agg_init<<<(N_NODES * HID + T - 1) / T, T, 0, stream>>>(bufA, dinv, b1, bufB);
  k_agg_edges<<<(N_EDGES * 32) / T, T, 0, stream>>>(bufA, src, dst, dinv, bufB);

  // conv2: t2 = relu(a1) @ W2 ; a2 = aggregate(t2) + b2
  k_gemm_wmma<HID, true><<<gemm_blocks, T, 0, stream>>>(bufB, W2, bufA, N_NODES);
  k_agg_init<<<(N_NODES * HID + T - 1) / T, T, 0, stream>>>(bufA, dinv, b2, bufB);
  k_agg_edges<<<(N_EDGES * 32) / T, T, 0, stream>>>(bufA, src, dst, dinv, bufB);

  // mean pool + head
  k_pool<<<(N_NODES * 32 + T - 1) / T, T, 0, stream>>>(bufB, batch, psum, pcnt);
  k_head<<<(N_GRAPHS + T - 1) / T, T, 0, stream>>>(psum, pcnt, gf, gW1, gb1, gW2, gb2,
                                                   cW1, cb1, cW2, cb2, out);
}